// GroupedStackedAFDF_41979010351591
// MI455X (gfx1250) — compile-verified
//
#include <hip/hip_runtime.h>

typedef __attribute__((ext_vector_type(2))) float v2f;
typedef __attribute__((ext_vector_type(8))) float v8f;

namespace {

constexpr int   kD     = 1024;
constexpr int   kL     = 4;
constexpr int   kG     = 32;
constexpr int   kRows  = 16;            // rows per workgroup (one WMMA M-tile)
constexpr int   kTP    = 33;            // padded 32x32 tile pitch (bank-conflict free)
constexpr int   kRP    = kG * kTP;      // 1056 floats per data row per component
constexpr int   kZHalf = kRows * kRP;   // 16896 floats (one component, one buffer)
constexpr int   kZBuf  = 2 * kZHalf;    // 33792 floats (re+im, one buffer)
constexpr int   kFOff  = 2 * kZBuf;     // 67584 floats (after both ping-pong buffers)
constexpr int   kSmemF = kFOff + 4 * 1024;  // + Fre/Fim/twre/twim tables
constexpr int   kSmemB = kSmemF * 4;        // 286720 bytes (< 320KB WGP LDS)
constexpr float kTwoPi = 6.28318530717958647692f;

__device__ __forceinline__ v8f wmma_f32(v2f a, v2f b, v8f c) {
  // D = A(16x4,f32) * B(4x16,f32) + C(16x16,f32)
  return __builtin_amdgcn_wmma_f32_16x16x4_f32(false, a, false, b, (short)0, c,
                                               false, false);
}

// One complex-matmul pass over the workgroup's 16-row x (32x32 complex tile) set.
//   BSRC: 0 = per-group global weight W[g][o][k]; 1 = DFT32 matrix; 2 = conj DFT32
//   TW:   0 = none; 1 = multiply output by e^{-2pi i (g*o)/1024}; 2 = conjugate twiddle
//   READ_ROW / WRITE_ROW: contract along / write to tile rows (true) or columns (false)
//   SCALE: multiply output by 1/1024 (IFFT normalization)
template <int BSRC, int TW, bool READ_ROW, bool WRITE_ROW, bool SCALE>
__device__ __forceinline__ void pass(const float* sRe, const float* sIm,
                                     float* dRe, float* dIm,
                                     const float* wRe, const float* wIm,
                                     const float* Fre, const float* Fim,
                                     const float* twre, const float* twim) {
  const int lane  = threadIdx.x & 31;
  const int wave  = threadIdx.x >> 5;
  const int mrow  = lane & 15;         // data row (WMMA M) in A fragment
  const int kh    = (lane >> 4) << 1;  // K sub-offset (0 or 2) per ISA A/B layout
  const int nlane = lane & 15;         // N index in B/D fragments
  const int mh    = lane >> 4;         // D row-half select

#pragma unroll 1
  for (int gi = 0; gi < 4; ++gi) {
    const int g = wave + gi * 8;       // 8 waves x 4 -> 32 groups

    // ---- A fragments: 16 rows x K=32 -> 8 WMMA K-steps ----
    v2f Ar[8], Ai[8];
#pragma unroll
    for (int s = 0; s < 8; ++s) {
      const int k0 = 4 * s + kh;
      if (READ_ROW) {
        const int off = mrow * kRP + g * kTP + k0;
        Ar[s].x = sRe[off];  Ar[s].y = sRe[off + 1];
        Ai[s].x = sIm[off];  Ai[s].y = sIm[off + 1];
      } else {
        const int off = mrow * kRP + k0 * kTP + g;
        Ar[s].x = sRe[off];  Ar[s].y = sRe[off + kTP];
        Ai[s].x = sIm[off];  Ai[s].y = sIm[off + kTP];
      }
    }

#pragma unroll
    for (int t = 0; t < 2; ++t) {
      const int o = t * 16 + nlane;    // output (N) index, 0..31 over two tiles

      // ---- B fragments: B[k][n] striped as k0/k0+1 per lane-half ----
      v2f Br[8], Bi[8];
#pragma unroll
      for (int s = 0; s < 8; ++s) {
        const int k0 = 4 * s + kh;
        if (BSRC == 0) {
          const int a = (g * 32 + o) * 32 + k0;   // W[g][o][k], k contiguous
          Br[s] = *(const v2f*)(wRe + a);
          Bi[s] = *(const v2f*)(wIm + a);
        } else {
          const int fi = o * 32 + k0;             // DFT32 is symmetric
          Br[s] = *(const v2f*)(Fre + fi);
          v2f fb = *(const v2f*)(Fim + fi);
          Bi[s] = (BSRC == 2) ? -fb : fb;
        }
      }

      // ---- complex GEMM via 4 independent WMMA accumulation chains ----
      v8f crp = {}, crm = {}, cip = {}, cim = {};
#pragma unroll
      for (int s = 0; s < 8; ++s) {
        crp = wmma_f32(Ar[s], Br[s], crp);   // Re*Re
        crm = wmma_f32(Ai[s], Bi[s], crm);   // Im*Im
        cip = wmma_f32(Ar[s], Bi[s], cip);   // Re*Im
        cim = wmma_f32(Ai[s], Br[s], cim);   // Im*Re
      }
      v8f cr = crp - crm;
      v8f ci = cip + cim;

      // ---- twiddle / scale / store back to LDS ----
      float wr = 1.0f, wi = 0.0f;
      if (TW != 0) {
        wr = twre[g * 32 + o];
        wi = (TW == 1) ? twim[g * 32 + o] : -twim[g * 32 + o];
      }
#pragma unroll
      for (int v = 0; v < 8; ++v) {
        const int m = v + 8 * mh;
        float rr = cr[v], ii = ci[v];
        if (TW != 0) {
          const float r2 = rr * wr - ii * wi;
          ii = rr * wi + ii * wr;
          rr = r2;
        }
        if (SCALE) { rr *= (1.0f / 1024.0f); ii *= (1.0f / 1024.0f); }
        const int off = WRITE_ROW ? (m * kRP + g * kTP + o)
                                  : (m * kRP + o * kTP + g);
        dRe[off] = rr;
        dIm[off] = ii;
      }
    }
  }
}

}  // namespace

__global__ void __launch_bounds__(256, 1)
afdf_fused_kernel(const float* __restrict__ x, const float* __restrict__ Aa,
                  const float* __restrict__ Ab, const float* __restrict__ Da,
                  const float* __restrict__ Db, const int* __restrict__ perms,
                  float* __restrict__ out) {
  extern __shared__ float smem[];
  // Ping-pong buffer pointers are computed at runtime (no LDS-pointer static
  // initializers — ld.lld cannot relocate addrspace(3) casts in initializers).
  auto bufRe = [&](int b) -> float* { return smem + b * kZBuf; };
  auto bufIm = [&](int b) -> float* { return smem + b * kZBuf + kZHalf; };
  float* Fre  = smem + kFOff;
  float* Fim  = smem + kFOff + 1024;
  float* twre = smem + kFOff + 2048;
  float* twim = smem + kFOff + 3072;

  const int  tid     = threadIdx.x;
  const long rowBase = (long)blockIdx.x * kRows;

  // DFT32 matrix F[p][q] = e^{-2pi i pq/32} and 1024-pt twiddles e^{-2pi i pq/1024}
  for (int i = tid; i < 1024; i += 256) {
    const int p = i >> 5, q = i & 31;
    const float aF = -kTwoPi * (float)((p * q) & 31) * (1.0f / 32.0f);
    Fre[i] = __cosf(aF);
    Fim[i] = __sinf(aF);
    const float aT = -kTwoPi * (float)(p * q) * (1.0f / 1024.0f);
    twre[i] = __cosf(aT);
    twim[i] = __sinf(aT);
  }

  // Stream input rows into LDS: j -> tile(j>>5, j&31), padded pitch 33
  for (int i = tid; i < kRows * kD; i += 256) {
    const int r = i >> 10, j = i & 1023;
    const float v = __builtin_nontemporal_load(x + (rowBase + r) * kD + j);
    const int off = r * kRP + (j >> 5) * kTP + (j & 31);
    bufRe(0)[off] = v;
    bufIm(0)[off] = 0.0f;
  }
  __syncthreads();

  int cur = 0;
#pragma unroll 1
  for (int l = 0; l < kL; ++l) {
    const float* waR = Aa + (size_t)l * (kG * 32 * 32);
    const float* waI = Ab + (size_t)l * (kG * 32 * 32);
    const float* wdR = Da + (size_t)l * (kG * 32 * 32);
    const float* wdI = Db + (size_t)l * (kG * 32 * 32);

    // 1) block-diagonal complex GEMM with Wa (rows -> rows)
    pass<0, 0, true, true, false>(bufRe(cur), bufIm(cur), bufRe(cur ^ 1), bufIm(cur ^ 1),
                                  waR, waI, Fre, Fim, twre, twim);
    __syncthreads(); cur ^= 1;
    // 2) FFT stage 1: contract strided index (cols), DFT32, apply twiddles
    pass<1, 1, false, false, false>(bufRe(cur), bufIm(cur), bufRe(cur ^ 1), bufIm(cur ^ 1),
                                    Fre, Fim, Fre, Fim, twre, twim);
    __syncthreads(); cur ^= 1;
    // 3) FFT stage 2: contract rows, DFT32, write canonical (cols)
    pass<1, 0, true, false, false>(bufRe(cur), bufIm(cur), bufRe(cur ^ 1), bufIm(cur ^ 1),
                                   Fre, Fim, Fre, Fim, twre, twim);
    __syncthreads(); cur ^= 1;
    // 4) block-diagonal complex GEMM with Wd
    pass<0, 0, true, true, false>(bufRe(cur), bufIm(cur), bufRe(cur ^ 1), bufIm(cur ^ 1),
                                  wdR, wdI, Fre, Fim, twre, twim);
    __syncthreads(); cur ^= 1;
    // 5) IFFT stage 1: conj DFT32, conj twiddles
    pass<2, 2, false, false, false>(bufRe(cur), bufIm(cur), bufRe(cur ^ 1), bufIm(cur ^ 1),
                                    Fre, Fim, Fre, Fim, twre, twim);
    __syncthreads(); cur ^= 1;
    // 6) IFFT stage 2: conj DFT32, scale 1/1024, write canonical
    pass<2, 0, true, false, true>(bufRe(cur), bufIm(cur), bufRe(cur ^ 1), bufIm(cur ^ 1),
                                  Fre, Fim, Fre, Fim, twre, twim);
    __syncthreads(); cur ^= 1;

    // 7) permutation: new[:, j] = old[:, perm[j]]
    const int* pm = perms + l * kD;
    {
      const float* sRe = bufRe(cur);
      const float* sIm = bufIm(cur);
      float* dRe = bufRe(cur ^ 1);
      float* dIm = bufIm(cur ^ 1);
      for (int i = tid; i < kRows * kD; i += 256) {
        const int r = i >> 10, j = i & 1023;
        const int pj = pm[j];
        const int so  = r * kRP + (pj >> 5) * kTP + (pj & 31);
        const int dof = r * kRP + (j >> 5) * kTP + (j & 31);
        dRe[dof] = sRe[so];
        dIm[dof] = sIm[so];
      }
    }
    __syncthreads(); cur ^= 1;
  }

  // Stream out the real part
  {
    const float* sRe = bufRe(cur);
    for (int i = tid; i < kRows * kD; i += 256) {
      const int r = i >> 10, j = i & 1023;
      const int off = r * kRP + (j >> 5) * kTP + (j & 31);
      __builtin_nontemporal_store(sRe[off], out + (rowBase + r) * kD + j);
    }
  }
}

extern "C" void kernel_launch(void* const* d_in, const int* in_sizes, int n_in,
                              void* d_out, int out_size, void* d_ws, size_t ws_size,
                              hipStream_t stream) {
  (void)n_in; (void)out_size; (void)d_ws; (void)ws_size;
  const float* x     = (const float*)d_in[0];
  const float* Aa    = (const float*)d_in[1];
  const float* Ab    = (const float*)d_in[2];
  const float* Da    = (const float*)d_in[3];
  const float* Db    = (const float*)d_in[4];
  const int*   perms = (const int*)d_in[5];
  float*       out   = (float*)d_out;

  const int nRows  = in_sizes[0] / kD;   // 32768
  const int blocks = nRows / kRows;      // 2048 workgroups x 16 rows

  // Allow 280KB dynamic LDS (gfx1250 WGP supports up to 320KB per workgroup).
  (void)hipFuncSetAttribute(reinterpret_cast<const void*>(&afdf_fused_kernel),
                            hipFuncAttributeMaxDynamicSharedMemorySize, kSmemB);

  afdf_fused_kernel<<<blocks, 256, kSmemB, stream>>>(x, Aa, Ab, Da, Db, perms, out);
}